// HashEncoding_44495861186613
// MI455X (gfx1250) — compile-verified
//
#include <hip/hip_runtime.h>
#include <stdint.h>

#ifndef __has_builtin
#define __has_builtin(x) 0
#endif

#define NUM_LEVELS   16
#define HASHMAP_SIZE (1u << 19)
#define HASH_MASK    (HASHMAP_SIZE - 1u)
#define BLOCK        256

typedef float v4f __attribute__((ext_vector_type(4)));
typedef int   v4i __attribute__((ext_vector_type(4)));
typedef __attribute__((address_space(1))) v4i gv4i_t;  // global int4*
typedef __attribute__((address_space(3))) v4i sv4i_t;  // LDS int4*

__device__ __forceinline__ void async_wait0() {
#if __has_builtin(__builtin_amdgcn_s_wait_asynccnt)
    __builtin_amdgcn_s_wait_asynccnt(0);
#else
    asm volatile("s_wait_asynccnt 0" ::: "memory");
#endif
}

__global__ __launch_bounds__(BLOCK) void hashenc_kernel(
    const float* __restrict__ x, const float* __restrict__ tables,
    float* __restrict__ out, int n)
{
    __shared__ __align__(16) float sx[BLOCK * 3];
    const int tid = threadIdx.x;
    const int p0  = blockIdx.x * BLOCK;
    const int pt  = p0 + tid;

    if (p0 + BLOCK <= n) {
        // Stage 256 points * 3 floats = 3072 B of coordinates into LDS,
        // fully coalesced 16B/lane, via the CDNA5 async global->LDS path.
        if (tid < (BLOCK * 3) / 4) {
            const float* gp = x + (size_t)p0 * 3 + (size_t)tid * 4;
#if __has_builtin(__builtin_amdgcn_global_load_async_to_lds_b128)
            __builtin_amdgcn_global_load_async_to_lds_b128(
                (gv4i_t*)gp, (sv4i_t*)&sx[tid * 4], 0, 0);
#else
            *(v4f*)&sx[tid * 4] = *(const v4f*)gp;
#endif
        }
#if __has_builtin(__builtin_amdgcn_global_load_async_to_lds_b128)
        async_wait0();
#endif
    } else if (pt < n) {  // tail block (not hit for N=2M, but stay correct)
        sx[tid * 3 + 0] = x[(size_t)pt * 3 + 0];
        sx[tid * 3 + 1] = x[(size_t)pt * 3 + 1];
        sx[tid * 3 + 2] = x[(size_t)pt * 3 + 2];
    }
    __syncthreads();

    if (pt >= n) return;

    const float CLIP_HI = 1.0f - 1e-6f;
    const float xn = fminf(fmaxf(sx[tid * 3 + 0], 0.0f), CLIP_HI);
    const float yn = fminf(fmaxf(sx[tid * 3 + 1], 0.0f), CLIP_HI);
    const float zn = fminf(fmaxf(sx[tid * 3 + 2], 0.0f), CLIP_HI);

    float acc[NUM_LEVELS * 2];

    // int(16 * 1.5**l) computed in double, truncated — matches Python int().
    constexpr int RES[NUM_LEVELS] = {16, 24, 36, 54, 81, 121, 182, 273,
                                     410, 615, 922, 1383, 2075, 3113, 4670, 7006};

    #pragma unroll
    for (int l = 0; l < NUM_LEVELS; ++l) {
        const float res = (float)RES[l];
        const float fx = xn * res, fy = yn * res, fz = zn * res;
        const float bx = floorf(fx), by = floorf(fy), bz = floorf(fz);
        const float wx1 = fx - bx, wy1 = fy - by, wz1 = fz - bz;
        const float wx0 = 1.0f - wx1, wy0 = 1.0f - wy1, wz0 = 1.0f - wz1;

        const uint32_t ix = (uint32_t)(int)bx;
        const uint32_t iy = (uint32_t)(int)by;
        const uint32_t iz = (uint32_t)(int)bz;
        // (c+1)*P == c*P + P : 3 muls + 3 adds instead of 24 muls
        const uint32_t hx0 = ix * 73856093u, hx1 = hx0 + 73856093u;
        const uint32_t hy0 = iy * 19349663u, hy1 = hy0 + 19349663u;
        const uint32_t hz0 = iz * 83492791u, hz1 = hz0 + 83492791u;

        const float2* __restrict__ tbl =
            (const float2*)tables + (size_t)l * HASHMAP_SIZE;

        // 8 gathers (global_load_b64, L2-resident: whole 64MB table set < 192MB L2)
        const float2 c000 = tbl[(hx0 ^ hy0 ^ hz0) & HASH_MASK];
        const float2 c001 = tbl[(hx0 ^ hy0 ^ hz1) & HASH_MASK];
        const float2 c010 = tbl[(hx0 ^ hy1 ^ hz0) & HASH_MASK];
        const float2 c011 = tbl[(hx0 ^ hy1 ^ hz1) & HASH_MASK];
        const float2 c100 = tbl[(hx1 ^ hy0 ^ hz0) & HASH_MASK];
        const float2 c101 = tbl[(hx1 ^ hy0 ^ hz1) & HASH_MASK];
        const float2 c110 = tbl[(hx1 ^ hy1 ^ hz0) & HASH_MASK];
        const float2 c111 = tbl[(hx1 ^ hy1 ^ hz1) & HASH_MASK];

        const float wxy00 = wx0 * wy0, wxy01 = wx0 * wy1;
        const float wxy10 = wx1 * wy0, wxy11 = wx1 * wy1;
        const float w000 = wxy00 * wz0, w001 = wxy00 * wz1;
        const float w010 = wxy01 * wz0, w011 = wxy01 * wz1;
        const float w100 = wxy10 * wz0, w101 = wxy10 * wz1;
        const float w110 = wxy11 * wz0, w111 = wxy11 * wz1;

        float f0 = w000 * c000.x;
        float f1 = w000 * c000.y;
        f0 = fmaf(w001, c001.x, f0);  f1 = fmaf(w001, c001.y, f1);
        f0 = fmaf(w010, c010.x, f0);  f1 = fmaf(w010, c010.y, f1);
        f0 = fmaf(w011, c011.x, f0);  f1 = fmaf(w011, c011.y, f1);
        f0 = fmaf(w100, c100.x, f0);  f1 = fmaf(w100, c100.y, f1);
        f0 = fmaf(w101, c101.x, f0);  f1 = fmaf(w101, c101.y, f1);
        f0 = fmaf(w110, c110.x, f0);  f1 = fmaf(w110, c110.y, f1);
        f0 = fmaf(w111, c111.x, f0);  f1 = fmaf(w111, c111.y, f1);

        acc[l * 2 + 0] = f0;
        acc[l * 2 + 1] = f1;
    }

    // 32 floats per point, contiguous: 8 non-temporal b128 stores.
    // A wave covers a contiguous 4KB region -> full-line write combining in GL2.
    float* op = out + (size_t)pt * (NUM_LEVELS * 2);
    #pragma unroll
    for (int i = 0; i < (NUM_LEVELS * 2) / 4; ++i) {
        v4f v = { acc[4 * i + 0], acc[4 * i + 1], acc[4 * i + 2], acc[4 * i + 3] };
        __builtin_nontemporal_store(v, (v4f*)(op + 4 * i));
    }
}

extern "C" void kernel_launch(void* const* d_in, const int* in_sizes, int n_in,
                              void* d_out, int out_size, void* d_ws, size_t ws_size,
                              hipStream_t stream) {
    const float* x      = (const float*)d_in[0];
    const float* tables = (const float*)d_in[1];
    float* out          = (float*)d_out;
    const int n    = in_sizes[0] / 3;
    const int grid = (n + BLOCK - 1) / BLOCK;
    hipLaunchKernelGGL(hashenc_kernel, dim3(grid), dim3(BLOCK), 0, stream,
                       x, tables, out, n);
}